// BCE_Loss_7164005449962
// MI455X (gfx1250) — compile-verified
//
#include <hip/hip_runtime.h>
#include <hip/hip_bf16.h>
#include <math.h>

// Problem constants (match reference)
#define B_N    8192
#define D_DIM  512
#define K_TOP  20

typedef float v2f __attribute__((ext_vector_type(2)));
typedef float v8f __attribute__((ext_vector_type(8)));

// ---------------------------------------------------------------------------
// Kernel 1: row L2-normalize (torch F.normalize: x / max(||x||, eps)).
// One 256-thread block per row; each thread handles a float2.
// Block 0 / thread 0 also zero-initializes the scalar output accumulator.
// ---------------------------------------------------------------------------
__global__ __launch_bounds__(256) void normalize_rows(const float* __restrict__ x,
                                                      float* __restrict__ xn,
                                                      float* __restrict__ out) {
  __shared__ float red[8];
  __shared__ float s_scale;
  const int row = blockIdx.x;
  const int t   = threadIdx.x;

  const float2* xr = reinterpret_cast<const float2*>(x + (size_t)row * D_DIM);
  float2 v = xr[t];
  float  ss = v.x * v.x + v.y * v.y;

  #pragma unroll
  for (int off = 16; off > 0; off >>= 1)
    ss += __shfl_down(ss, off, 32);
  if ((t & 31) == 0) red[t >> 5] = ss;
  __syncthreads();

  if (t == 0) {
    float s = 0.0f;
    #pragma unroll
    for (int i = 0; i < 8; ++i) s += red[i];
    float nrm = fmaxf(sqrtf(s), 1e-12f);
    s_scale = 1.0f / nrm;
    if (row == 0) out[0] = 0.0f;   // zero the loss accumulator (stream-ordered)
  }
  __syncthreads();

  const float sc = s_scale;
  float2 o;
  o.x = v.x * sc;
  o.y = v.y * sc;
  reinterpret_cast<float2*>(xn + (size_t)row * D_DIM)[t] = o;
}

// ---------------------------------------------------------------------------
// Kernel 2: fused cos = xn @ xn^T -> per-row top-20 (diag excluded) -> BCE.
// Block = 256 threads = 8 waves; block owns 16 rows (i0..i0+15).
// Each wave computes a 16x64 strip per pass: 4 C tiles register-blocked in N,
// so one A-fragment load feeds 4 v_wmma_f32_16x16x4_f32 ops
// (1.25 global_load_b64 per WMMA instead of 2).
//
// Fragment layout (wave32, 16x16x4 f32):
//   A (16x4): lane = M (mod 16); v0/v1 hold K = {0,1} (lanes 0-15) or {2,3}
//   B (4x16): lane = N (mod 16); same K split. Since B = xn^T, both frags are
//   the identical gather: xn[rowOrCol*512 + kk + 2*(lane>>4) + {0,1}].
// C layout: lane L, vgpr v -> row = v + 8*(L>>4), col = L&15.
// ---------------------------------------------------------------------------
__global__ __launch_bounds__(256) void knn_bce(const float* __restrict__ xn,
                                               const int* __restrict__ labels,
                                               float* __restrict__ out) {
  __shared__ float s_tile[8][16][16];        // per-wave C-tile transpose scratch
  __shared__ float s_val[256][K_TOP];        // per-lane top-k values
  __shared__ int   s_idx[256][K_TOP];        // per-lane top-k column indices

  const int tid  = threadIdx.x;
  const int wave = tid >> 5;
  const int lane = tid & 31;
  const int r16  = lane & 15;
  const int half = lane >> 4;                // K-pair select / column half
  const int i0   = blockIdx.x * 16;
  const int rowA = i0 + r16;

  // init this lane's top-k list (cos >= -1 > -2 sentinel, so lists always fill)
  #pragma unroll
  for (int p = 0; p < K_TOP; ++p) {
    s_val[tid][p] = -2.0f;
    s_idx[tid][p] = -1;
  }
  float tmin = -2.0f;   // cached min of this lane's list
  int   tpos = 0;       // its position

  const float* aBase = xn + (size_t)rowA * D_DIM + 2 * half;

  // 128 groups of 64 columns, round-robined over 8 waves -> 16 groups/wave
  for (int g = wave; g < (B_N / 64); g += 8) {
    const int j0 = g * 64;
    const float* bBase = xn + (size_t)(j0 + r16) * D_DIM + 2 * half;

    v8f cs[4];
    #pragma unroll
    for (int t = 0; t < 4; ++t) cs[t] = (v8f){};

    #pragma unroll 4
    for (int kk = 0; kk < D_DIM; kk += 4) {
      v2f a = *reinterpret_cast<const v2f*>(aBase + kk);
      #pragma unroll
      for (int t = 0; t < 4; ++t) {
        v2f b = *reinterpret_cast<const v2f*>(bBase + (size_t)t * 16 * D_DIM + kk);
        cs[t] = __builtin_amdgcn_wmma_f32_16x16x4_f32(false, a, false, b,
                                                      (short)0, cs[t],
                                                      false, false);
      }
    }

    // Drain the 4 sub-tiles through the per-wave LDS scratch sequentially.
    // Same-wave DS ops are in-order, so tile t+1 stores cannot pass tile t
    // loads, and loads below observe the stores above.
    #pragma unroll
    for (int t = 0; t < 4; ++t) {
      const int jt0 = j0 + t * 16;
      // scatter: element (row = v + 8*half, col = r16)
      #pragma unroll
      for (int v = 0; v < 8; ++v)
        s_tile[wave][v + 8 * half][r16] = cs[t][v];
      // gather: this lane consumes row r16, columns [half*8, half*8+8)
      #pragma unroll
      for (int cc = 0; cc < 8; ++cc) {
        const int   col = half * 8 + cc;
        const float val = s_tile[wave][r16][col];
        const int   j   = jt0 + col;
        // diagonal exclusion == reference's fill_diagonal(min-1) for top-k
        if (j != rowA && val > tmin) {
          s_val[tid][tpos] = val;
          s_idx[tid][tpos] = j;
          float m  = s_val[tid][0];
          int   mp = 0;
          #pragma unroll
          for (int p = 1; p < K_TOP; ++p) {
            float vv = s_val[tid][p];
            if (vv < m) { m = vv; mp = p; }
          }
          tmin = m;
          tpos = mp;
        }
      }
    }
  }

  __syncthreads();

  // Merge the 16 partial lists per row (8 waves x 2 column-halves) and
  // accumulate the BCE terms. Thread r exclusively owns lists with lane%16==r.
  if (tid < 16) {
    const int r     = tid;
    const int myLab = labels[i0 + r];
    float     lsum  = 0.0f;
    for (int pick = 0; pick < K_TOP; ++pick) {
      float best = -10.0f;
      int   bl = 0, bp = 0;
      for (int w = 0; w < 16; ++w) {
        const int li = ((w >> 1) * 32) + ((w & 1) * 16) + r;
        #pragma unroll 4
        for (int p = 0; p < K_TOP; ++p) {
          const float v = s_val[li][p];
          if (v > best) { best = v; bl = li; bp = p; }
        }
      }
      s_val[bl][bp] = -10.0f;                 // remove from candidate pool
      const int   j    = s_idx[bl][bp];
      const float pred = (best + 1.0f) * 0.5f;
      const bool  tl   = (labels[j] == myLab);
      const float logp   = fmaxf(logf(pred),    -100.0f);
      const float log1mp = fmaxf(log1pf(-pred), -100.0f);
      lsum += -(tl ? logp : log1mp);
    }
    atomicAdd(out, lsum * (1.0f / ((float)B_N * (float)K_TOP)));
  }
}

// ---------------------------------------------------------------------------
extern "C" void kernel_launch(void* const* d_in, const int* in_sizes, int n_in,
                              void* d_out, int out_size, void* d_ws, size_t ws_size,
                              hipStream_t stream) {
  (void)in_sizes; (void)n_in; (void)out_size; (void)ws_size;
  const float* batch  = (const float*)d_in[0];
  const int*   labels = (const int*)d_in[1];
  float* out = (float*)d_out;
  float* xn  = (float*)d_ws;   // 8192*512 fp32 = 16 MB of workspace

  normalize_rows<<<B_N, 256, 0, stream>>>(batch, xn, out);
  knn_bce<<<B_N / 16, 256, 0, stream>>>(xn, labels, out);
}